// DeepResidualGCN_63024350102324
// MI455X (gfx1250) — compile-verified
//
#include <hip/hip_runtime.h>

typedef __attribute__((ext_vector_type(2))) float v2f;
typedef __attribute__((ext_vector_type(8))) float v8f;

#define N_NODES 100000
#define N_EDGES 1600000
#define DIM     128
#define NLAYERS 8

// ---------------- utility kernels ----------------

__global__ __launch_bounds__(256) void k_zero4(float4* p, int n4) {
  int i = blockIdx.x * 256 + threadIdx.x;
  if (i < n4) p[i] = make_float4(0.f, 0.f, 0.f, 0.f);
}

__global__ __launch_bounds__(256) void k_deg(const int* __restrict__ dst,
                                             float* __restrict__ deg, int E) {
  int e = blockIdx.x * 256 + threadIdx.x;
  if (e < E)
    __hip_atomic_fetch_add(&deg[dst[e]], 1.0f, __ATOMIC_RELAXED,
                           __HIP_MEMORY_SCOPE_AGENT);
}

__global__ __launch_bounds__(256) void k_dis(float* deg, int n) {
  int i = blockIdx.x * 256 + threadIdx.x;
  if (i < n) deg[i] = rsqrtf(deg[i] + 1.0f);
}

__global__ __launch_bounds__(256) void k_norm(const int* __restrict__ src,
                                              const int* __restrict__ dst,
                                              const float* __restrict__ dis,
                                              float* __restrict__ norm, int E) {
  int e = blockIdx.x * 256 + threadIdx.x;
  if (e < E) norm[e] = dis[src[e]] * dis[dst[e]];
}

// ---------------- WMMA f32 GEMM: Out[N,128] = A[N,128] @ W[128,128] ----------------
// One wave computes a 16-row x 128-col stripe.  W staged in LDS (64 KB).
// A fragment (16x4 f32): lanes 0-15 -> M=0..15 (K = k0,k0+1), lanes 16-31 -> K = k0+2,k0+3.
// B fragment (4x16 f32): lane%16 -> N, VGPR v -> K = k0 + 2*(lane/16) + v.
// C/D (16x16 f32): VGPR r -> M = r + 8*(lane/16), N = lane%16.

__global__ __launch_bounds__(256) void k_gemm(const float* __restrict__ A,
                                              const float* __restrict__ Wg,
                                              float* __restrict__ Out, int nrows) {
  __shared__ float Wl[DIM * DIM];
  for (int i = threadIdx.x; i < DIM * DIM / 4; i += 256)
    ((float4*)Wl)[i] = ((const float4*)Wg)[i];
  __syncthreads();

  const int wid  = threadIdx.x >> 5;
  const int lane = threadIdx.x & 31;
  const int mt   = blockIdx.x * 8 + wid;
  if (mt >= nrows / 16) return;           // uniform per wave -> EXEC all-ones for WMMA
  const int half = lane >> 4;
  const int l16  = lane & 15;
  const size_t rowbase = (size_t)(mt * 16 + l16) * DIM;

  v8f acc[8] = {};
  for (int k0 = 0; k0 < DIM; k0 += 4) {
    const int ka = k0 + 2 * half;
    v2f a = *(const v2f*)(A + rowbase + ka);
#pragma unroll
    for (int t = 0; t < 8; t++) {
      const int col = t * 16 + l16;
      v2f b;
      b.x = Wl[ka * DIM + col];
      b.y = Wl[(ka + 1) * DIM + col];
      acc[t] = __builtin_amdgcn_wmma_f32_16x16x4_f32(
          false, a, false, b, (short)0, acc[t], false, false);
    }
  }

  const size_t outbase = (size_t)mt * 16 * DIM;
#pragma unroll
  for (int t = 0; t < 8; t++) {
    const int col = t * 16 + l16;
#pragma unroll
    for (int r = 0; r < 8; r++)
      Out[outbase + (size_t)(r + 8 * half) * DIM + col] = acc[t][r];
  }
}

// ---------------- edge scatter: agg[dst] += hW[src] * norm[e] ----------------
// One wave per edge: lane reads float4 (coalesced 512B row), 4 relaxed f32 atomics.

__global__ __launch_bounds__(256) void k_scatter(const float* __restrict__ hW,
                                                 const int* __restrict__ src,
                                                 const int* __restrict__ dst,
                                                 const float* __restrict__ norm,
                                                 float* __restrict__ agg, int E) {
  const long long t = (long long)blockIdx.x * 256 + threadIdx.x;
  const int e = (int)(t >> 5);
  if (e >= E) return;
  const int lane = (int)(t & 31);
  const int s = src[e], d = dst[e];
  const float nv = norm[e];
  const float4 v = *(const float4*)(hW + (size_t)s * DIM + lane * 4);
  float* o = agg + (size_t)d * DIM + lane * 4;
  __hip_atomic_fetch_add(o + 0, v.x * nv, __ATOMIC_RELAXED, __HIP_MEMORY_SCOPE_AGENT);
  __hip_atomic_fetch_add(o + 1, v.y * nv, __ATOMIC_RELAXED, __HIP_MEMORY_SCOPE_AGENT);
  __hip_atomic_fetch_add(o + 2, v.z * nv, __ATOMIC_RELAXED, __HIP_MEMORY_SCOPE_AGENT);
  __hip_atomic_fetch_add(o + 3, v.w * nv, __ATOMIC_RELAXED, __HIP_MEMORY_SCOPE_AGENT);
}

// ---------------- finalize kernels ----------------
// finA: v = agg + hW*dis^2 + b (+ prev_initial);  [initial = v];  h = relu(v)
// finB: h += agg + hW*dis^2 + b        (second conv of the layer)

__global__ __launch_bounds__(256) void k_finA(const float* __restrict__ agg,
                                              const float* __restrict__ hW,
                                              const float* __restrict__ dis,
                                              const float* __restrict__ bias,
                                              const float* __restrict__ prev,
                                              float* __restrict__ init_out,
                                              float* __restrict__ hout, int n) {
  const long long t = (long long)blockIdx.x * 256 + threadIdx.x;
  const int node = (int)(t >> 5);
  if (node >= n) return;
  const int lane = (int)(t & 31);
  const float dv = dis[node];
  const float sn = dv * dv;
  const size_t off = (size_t)node * DIM + lane * 4;
  float4 a = *(const float4*)(agg + off);
  float4 w = *(const float4*)(hW + off);
  float4 b = ((const float4*)bias)[lane];
  float4 v;
  v.x = a.x + w.x * sn + b.x;
  v.y = a.y + w.y * sn + b.y;
  v.z = a.z + w.z * sn + b.z;
  v.w = a.w + w.w * sn + b.w;
  if (prev) {
    float4 p = *(const float4*)(prev + off);
    v.x += p.x; v.y += p.y; v.z += p.z; v.w += p.w;
  }
  if (init_out) *(float4*)(init_out + off) = v;
  float4 r;
  r.x = fmaxf(v.x, 0.f); r.y = fmaxf(v.y, 0.f);
  r.z = fmaxf(v.z, 0.f); r.w = fmaxf(v.w, 0.f);
  *(float4*)(hout + off) = r;
}

__global__ __launch_bounds__(256) void k_finB(const float* __restrict__ agg,
                                              const float* __restrict__ hW,
                                              const float* __restrict__ dis,
                                              const float* __restrict__ bias,
                                              float* __restrict__ h, int n) {
  const long long t = (long long)blockIdx.x * 256 + threadIdx.x;
  const int node = (int)(t >> 5);
  if (node >= n) return;
  const int lane = (int)(t & 31);
  const float dv = dis[node];
  const float sn = dv * dv;
  const size_t off = (size_t)node * DIM + lane * 4;
  float4 a = *(const float4*)(agg + off);
  float4 w = *(const float4*)(hW + off);
  float4 b = ((const float4*)bias)[lane];
  float4 o = *(const float4*)(h + off);
  o.x += a.x + w.x * sn + b.x;
  o.y += a.y + w.y * sn + b.y;
  o.z += a.z + w.z * sn + b.z;
  o.w += a.w + w.w * sn + b.w;
  *(float4*)(h + off) = o;
}

// ---------------- host-side orchestration ----------------

extern "C" void kernel_launch(void* const* d_in, const int* in_sizes, int n_in,
                              void* d_out, int out_size, void* d_ws, size_t ws_size,
                              hipStream_t stream) {
  const float* x  = (const float*)d_in[0];           // [N,128]
  const int*   ei = (const int*)d_in[1];             // [2,E] int32
  const float* W  = (const float*)d_in[2];           // [L,128,128]
  const float* B  = (const float*)d_in[3];           // [L,128]
  float* out = (float*)d_out;                        // [N,128]

  const int* src = ei;
  const int* dst = ei + N_EDGES;

  // carve workspace
  char* w = (char*)d_ws;
  auto carve = [&](size_t bytes) {
    float* p = (float*)w;
    w += ((bytes + 255) / 256) * 256;
    return p;
  };
  float* dis  = carve((size_t)N_NODES * 4);
  float* nrm  = carve((size_t)N_EDGES * 4);
  float* h    = carve((size_t)N_NODES * DIM * 4);
  float* init = carve((size_t)N_NODES * DIM * 4);
  float* hW   = carve((size_t)N_NODES * DIM * 4);
  float* agg  = carve((size_t)N_NODES * DIM * 4);

  const int blkE    = (N_EDGES + 255) / 256;              // 6250
  const int blkN    = (N_NODES + 255) / 256;              // 391
  const int blkGemm = (N_NODES / 16 + 7) / 8;             // 782
  const int blkScat = (int)(((long long)N_EDGES * 32) / 256);  // 200000
  const int blkFin  = (N_NODES * 32) / 256;               // 12500
  const int blkZagg = (N_NODES * DIM / 4 + 255) / 256;    // 12500
  const int blkZdis = (N_NODES / 4 + 255) / 256;          // 98

  // degree -> dis -> per-edge norm
  k_zero4<<<blkZdis, 256, 0, stream>>>((float4*)dis, N_NODES / 4);
  k_deg<<<blkE, 256, 0, stream>>>(dst, dis, N_EDGES);
  k_dis<<<blkN, 256, 0, stream>>>(dis, N_NODES);
  k_norm<<<blkE, 256, 0, stream>>>(src, dst, dis, nrm, N_EDGES);

  auto conv = [&](const float* hin, int layer) {
    k_gemm<<<blkGemm, 256, 0, stream>>>(hin, W + (size_t)layer * DIM * DIM, hW, N_NODES);
    k_zero4<<<blkZagg, 256, 0, stream>>>((float4*)agg, N_NODES * DIM / 4);
    k_scatter<<<blkScat, 256, 0, stream>>>(hW, src, dst, nrm, agg, N_EDGES);
  };

  for (int i = 0; i < NLAYERS; i++) {
    const float* hin = (i == 0) ? x : h;
    conv(hin, i);
    const float* prev = (i == 0) ? nullptr : ((i == 1) ? x : init);
    float* init_out   = (i == 0) ? nullptr : init;
    float* hout       = (i == NLAYERS - 1) ? out : h;
    k_finA<<<blkFin, 256, 0, stream>>>(agg, hW, dis, B + i * DIM, prev, init_out,
                                       hout, N_NODES);
    if (i < NLAYERS - 1) {
      const int j = (i == 0) ? (NLAYERS - 1) : (i - 1);   // Ws[-1] quirk at i==0
      conv(h, j);
      k_finB<<<blkFin, 256, 0, stream>>>(agg, hW, dis, B + j * DIM, h, N_NODES);
    }
  }
}